// LapLoss_63161789055589
// MI455X (gfx1250) — compile-verified
//
#include <hip/hip_runtime.h>

// Laplacian-pyramid L1 loss, MI455X (gfx1250), wave32.
// loss = sum_l mean |LapPyr(input)[l] - LapPyr(target)[l]|
// Linearity: == sum_l mean |LapPyr(input - target)[l]|  -> single pyramid.

#define LEVELS 5
#define Bn 32
#define Cn 3

typedef float v2f __attribute__((ext_vector_type(2)));
typedef float v8f __attribute__((ext_vector_type(8)));

__device__ __forceinline__ int refl(int p, int n) {
    p = p < 0 ? -p : p;
    return p >= n ? 2 * n - 2 - p : p;
}

// Full-wave (32 lane) sum using V_WMMA_F32_16X16X4_F32.
// A(16x4): lane<16 holds (K0,K1)=(v,0) for M=lane; lane>=16 holds (K2,K3)=(v,0) for M=lane-16.
// B(4x16) = ones. C[m,n] = v_m + v_{m+16} (independent of n).
// Lane l<16 sees rows 0..7 in c[0..7]; lane l>=16 sees rows 8..15.
// Local sum of c[0..7] + shfl_xor(16) => total in every lane. Exact f32 adds.
__device__ __forceinline__ float wave_sum_wmma(float v) {
    v2f a; a[0] = v;    a[1] = 0.0f;
    v2f b; b[0] = 1.0f; b[1] = 1.0f;
    v8f c = {};
    c = __builtin_amdgcn_wmma_f32_16x16x4_f32(false, a, false, b,
                                              (short)0, c, false, false);
    float s = c[0] + c[1] + c[2] + c[3] + c[4] + c[5] + c[6] + c[7];
    s += __shfl_xor(s, 16, 32);
    return s;
}

// One pyramid level, fully fused:
//   cur (H x W)  --conv5x5+stride2-->  down (Hc x Wc)   [stored for next level]
//   down --zero-stuff+conv(4k)--> up ; lap = cur - up ; partial[block] = sum |lap|
// Block: 16x16 threads, one 32x32 fine tile.
__global__ __launch_bounds__(256)
void lap_level_kernel(const float* __restrict__ curA,
                      const float* __restrict__ curB,   // target (level 0 only)
                      const float* __restrict__ kern,   // (C,1,5,5)
                      float* __restrict__ down,         // (B,C,Hc,Wc)
                      float* __restrict__ partial,      // one slot per block
                      int H, int W, int isL0)
{
    const int Hc = H >> 1, Wc = W >> 1;
    const int bc  = blockIdx.z;            // b*C + c
    const int ch  = bc % Cn;
    const int ty  = threadIdx.y, tx = threadIdx.x;
    const int tid = ty * 16 + tx;
    const int yt  = blockIdx.y * 32, xt = blockIdx.x * 32;
    const int Yt0 = yt >> 1,        Xt0 = xt >> 1;

    __shared__ float fine[40][41];     // fine tile with 4-wide halo (reflect applied)
    __shared__ float coarse[18][19];   // down tile with 1-ring halo
    __shared__ float skern[25];
    __shared__ float wsum[8];

    if (tid < 25) skern[tid] = kern[ch * 25 + tid];

    const float* baseA = curA + (size_t)bc * H * W;
    const float* baseB = curB + (size_t)bc * H * W;

    // Load 40x40 fine tile (reflect padded); level 0 computes e = input - target.
    for (int idx = tid; idx < 1600; idx += 256) {
        int ly = idx / 40, lx = idx - ly * 40;
        int gy = refl(yt - 4 + ly, H);
        int gx = refl(xt - 4 + lx, W);
        float v = baseA[(size_t)gy * W + gx];
        if (isL0) v -= baseB[(size_t)gy * W + gx];
        fine[ly][lx] = v;
    }
    __syncthreads();

    // 18x18 coarse tile: down(Yc) = sum_{i,j} k[i][j] * fineP(2*Yc+i-2, 2*Xc+j-2)
    // fine rel index = 2*cy + i (origin yt-4).
    for (int idx = tid; idx < 324; idx += 256) {
        int cy = idx / 18, cx = idx - cy * 18;
        float acc = 0.0f;
#pragma unroll
        for (int i = 0; i < 5; ++i)
#pragma unroll
            for (int j = 0; j < 5; ++j)
                acc += skern[i * 5 + j] * fine[2 * cy + i][2 * cx + j];
        coarse[cy][cx] = acc;
    }
    __syncthreads();

    // Store central 16x16 of coarse tile -> down buffer (exactly one writer per elem).
    {
        int Y = Yt0 + ty, X = Xt0 + tx;
        down[((size_t)bc * Hc + Y) * Wc + X] = coarse[ty + 1][tx + 1];
    }

    // Upsample (zero-stuffed conv with 4*k) + |cur - up|; 2x2 fine pixels per thread.
    float lsum = 0.0f;
#pragma unroll
    for (int sy = 0; sy < 2; ++sy) {
#pragma unroll
        for (int sx = 0; sx < 2; ++sx) {
            float up = 0.0f;
#pragma unroll
            for (int dy = -2; dy <= 2; ++dy) {
                if (((dy + sy) & 1) != 0) continue;      // stuffed row must be even
                int cy = ty + ((sy + dy) >> 1) + 1;      // 0..17
#pragma unroll
                for (int dx = -2; dx <= 2; ++dx) {
                    if (((dx + sx) & 1) != 0) continue;
                    int cx = tx + ((sx + dx) >> 1) + 1;
                    up += skern[(dy + 2) * 5 + (dx + 2)] * coarse[cy][cx];
                }
            }
            up *= 4.0f;
            float cv = fine[2 * ty + sy + 4][2 * tx + sx + 4];
            lsum += fabsf(cv - up);
        }
    }

    // Block reduction: WMMA wave-sum (all 32 lanes active) + 8 wave partials.
    float ws = wave_sum_wmma(lsum);
    if ((tid & 31) == 0) wsum[tid >> 5] = ws;
    __syncthreads();
    if (tid == 0) {
        float s = 0.0f;
#pragma unroll
        for (int w = 0; w < 8; ++w) s += wsum[w];
        int blin = (blockIdx.z * gridDim.y + blockIdx.y) * gridDim.x + blockIdx.x;
        partial[blin] = s;   // deterministic: unique slot per block
    }
}

// Reduce all per-block partials into the final scalar loss (single block).
__global__ __launch_bounds__(256)
void combine_kernel(const float* __restrict__ partial, float* __restrict__ out)
{
    __shared__ float wsum[8];
    __shared__ float lvl[LEVELS];
    const int tid = threadIdx.x;

    int base = 0;
    for (int l = 0; l < LEVELS; ++l) {
        int t   = (512 >> l) / 32;
        int cnt = Bn * Cn * t * t;          // blocks at this level
        float v = 0.0f;
        for (int i = tid; i < cnt; i += 256) v += partial[base + i];
        float ws = wave_sum_wmma(v);
        if ((tid & 31) == 0) wsum[tid >> 5] = ws;
        __syncthreads();
        if (tid == 0) {
            float s = 0.0f;
#pragma unroll
            for (int w = 0; w < 8; ++w) s += wsum[w];
            lvl[l] = s;
        }
        __syncthreads();
        base += cnt;
    }

    if (tid == 0) {
        float total = 0.0f;
        for (int l = 0; l < LEVELS; ++l) {
            int   Hl = 512 >> l;
            float n  = (float)Bn * (float)Cn * (float)Hl * (float)Hl;
            total += lvl[l] / n;
        }
        out[0] = total;
    }
}

extern "C" void kernel_launch(void* const* d_in, const int* in_sizes, int n_in,
                              void* d_out, int out_size, void* d_ws, size_t ws_size,
                              hipStream_t stream)
{
    const float* a    = (const float*)d_in[0];   // input  (32,3,512,512) f32
    const float* b    = (const float*)d_in[1];   // target (32,3,512,512) f32
    const float* kern = (const float*)d_in[2];   // kernel (3,1,5,5) f32
    float* ws = (float*)d_ws;

    // Workspace layout: down buffers for levels 0..4, then per-block partials.
    size_t off[LEVELS];
    size_t o = 0;
    for (int l = 0; l < LEVELS; ++l) {
        off[l] = o;
        int Hc = 256 >> l;
        o += (size_t)Bn * Cn * Hc * Hc;          // 8,380,416 floats total
    }
    float* partial = ws + o;                      // 32,736 floats total

    size_t pbase = 0;
    for (int l = 0; l < LEVELS; ++l) {
        int H = 512 >> l;
        int t = H / 32;
        dim3 grid(t, t, Bn * Cn);
        dim3 blk(16, 16);
        lap_level_kernel<<<grid, blk, 0, stream>>>(
            (l == 0) ? a : (ws + off[l - 1]),
            b, kern,
            ws + off[l],
            partial + pbase,
            H, H, (l == 0) ? 1 : 0);
        pbase += (size_t)Bn * Cn * t * t;
    }

    combine_kernel<<<1, 256, 0, stream>>>(partial, (float*)d_out);
}